// BidirectionalRNN_43121471651930
// MI455X (gfx1250) — compile-verified
//
#include <hip/hip_runtime.h>

// Bidirectional 2-layer tanh RNN on gfx1250 using exact-fp32 WMMA
// (v_wmma_f32_16x16x4_f32) for both the input projections and the recurrent
// h @ Wh^T step, with a persistent-kernel per-step device barrier.

typedef __attribute__((ext_vector_type(2))) float v2f;
typedef __attribute__((ext_vector_type(8))) float v8f;

#define BATCH 64
#define SEQ   512
#define HID   512
#define NBLK  32   // blocks per direction in the scan kernel

__device__ __forceinline__ v8f wmma_f32_k4(v2f a, v2f b, v8f c) {
  // 8 args: (neg_a, A, neg_b, B, c_mod, C, reuse_a, reuse_b)
  return __builtin_amdgcn_wmma_f32_16x16x4_f32(false, a, false, b, (short)0, c,
                                               false, false);
}

// ---------------------------------------------------------------------------
// C[M,N] = A[M,K] * W[N,K]^T + bias[N]
// Block: 256 threads (8 waves), 64x64 C tile, 32-wide K chunks in LDS.
// Wave w: m-tile = w&3, n-half = w>>2 (two 16x16 accumulators).
// ---------------------------------------------------------------------------
__global__ __launch_bounds__(256)
void proj_gemm(const float* __restrict__ A, const float* __restrict__ W,
               const float* __restrict__ bias, float* __restrict__ C,
               int M, int N, int K) {
  __shared__ float As[64][34];  // stride 34: even (8B-aligned v2f) + conflict-free
  __shared__ float Ws[64][34];

  const int tid  = threadIdx.x;
  const int wave = tid >> 5;
  const int lane = tid & 31;
  const int l16  = lane & 15;
  const int khi  = (lane >> 4) << 1;  // 0 for lanes 0-15, 2 for lanes 16-31
  const int wr   = wave & 3;
  const int wc   = wave >> 2;
  const int m0   = blockIdx.y * 64;
  const int n0   = blockIdx.x * 64;

  v8f acc0 = {};
  v8f acc1 = {};

  for (int k0 = 0; k0 < K; k0 += 32) {
    for (int i = tid; i < 64 * 32; i += 256) {
      const int r = i >> 5, c = i & 31;
      As[r][c] = A[(size_t)(m0 + r) * K + k0 + c];
      Ws[r][c] = W[(size_t)(n0 + r) * K + k0 + c];
    }
    __syncthreads();
#pragma unroll
    for (int kk = 0; kk < 32; kk += 4) {
      const v2f a  = *(const v2f*)&As[wr * 16 + l16][kk + khi];
      const v2f b0 = *(const v2f*)&Ws[wc * 32 + l16][kk + khi];
      const v2f b1 = *(const v2f*)&Ws[wc * 32 + 16 + l16][kk + khi];
      acc0 = wmma_f32_k4(a, b0, acc0);
      acc1 = wmma_f32_k4(a, b1, acc1);
    }
    __syncthreads();
  }

  const int col0 = n0 + wc * 32 + l16;
  const int col1 = col0 + 16;
  const float bi0 = bias[col0];
  const float bi1 = bias[col1];
#pragma unroll
  for (int i = 0; i < 8; ++i) {
    const int row = m0 + wr * 16 + ((lane < 16) ? i : (8 + i));
    C[(size_t)row * N + col0] = acc0[i] + bi0;
    C[(size_t)row * N + col1] = acc1[i] + bi1;
  }
}

// ---------------------------------------------------------------------------
// Zero h double-buffers and per-step barrier counters (re-run every launch
// so graph replay is deterministic).
// ---------------------------------------------------------------------------
__global__ void init_state(float* HbufF, float* HbufB, int* ctr) {
  const int i = blockIdx.x * blockDim.x + threadIdx.x;
  if (i < 2 * BATCH * HID) { HbufF[i] = 0.0f; HbufB[i] = 0.0f; }
  if (i < 2 * SEQ) ctr[i] = 0;
}

// ---------------------------------------------------------------------------
// Persistent scan: h_t = tanh(pre[:,t] + h_prev @ Wh^T + bh), both directions.
// 64 blocks x 128 threads: blocks 0..31 = forward, 32..63 = backward.
// Block blk owns 16 output columns (n0 = blk*16); its Wh rows live in LDS for
// the whole kernel. Each of its 4 waves owns one 16-row batch tile.
// Per step: chunked h[64,32] staged in LDS -> 8 f32 WMMAs per chunk per wave.
// Steps are separated by a per-direction device-wide atomic barrier.
// ---------------------------------------------------------------------------
__global__ __launch_bounds__(128)
void rnn_scan(const float* __restrict__ preF, const float* __restrict__ preB,
              const float* __restrict__ WhF,  const float* __restrict__ WhB,
              const float* __restrict__ bhF,  const float* __restrict__ bhB,
              float* __restrict__ out,        // [B][S][2H]
              float* HbufF, float* HbufB,     // [2][B][H] each
              int* ctrF, int* ctrB,           // [S] each
              float* hnF, float* hnB) {       // [B][H] each
  __shared__ float Whs[16][514];  // stride 514: 8B-aligned, bank-spread
  __shared__ float Hk[64][34];

  const int bb  = blockIdx.x;
  const int dir = bb >> 5;
  const int blk = bb & (NBLK - 1);

  const float* pre  = dir ? preB  : preF;
  const float* Wh   = dir ? WhB   : WhF;
  const float* bh   = dir ? bhB   : bhF;
  float*       Hbuf = dir ? HbufB : HbufF;
  int*         ctr  = dir ? ctrB  : ctrF;
  float*       hn   = dir ? hnB   : hnF;

  const int tid  = threadIdx.x;
  const int wave = tid >> 5;          // 0..3 : batch tile
  const int lane = tid & 31;
  const int l16  = lane & 15;
  const int khi  = (lane >> 4) << 1;
  const int n0   = blk * 16;

  // Pin this block's 16 rows of Wh (constant across steps) in LDS.
  for (int i = tid; i < 16 * HID; i += 128) {
    const int r = i >> 9, c = i & (HID - 1);
    Whs[r][c] = Wh[(size_t)(n0 + r) * HID + c];
  }
  const int   col  = n0 + l16;
  const float bias = bh[col];
  __syncthreads();

  for (int step = 0; step < SEQ; ++step) {
    const int tt = dir ? (SEQ - 1 - step) : step;
    const float* Hprev = Hbuf + (size_t)(step & 1) * BATCH * HID;
    float*       Hnext = Hbuf + (size_t)((step + 1) & 1) * BATCH * HID;

    v8f acc = {};
    for (int k0 = 0; k0 < HID; k0 += 32) {
      for (int i = tid; i < 64 * 32; i += 128) {
        const int r = i >> 5, c = i & 31;
        Hk[r][c] = Hprev[r * HID + k0 + c];
      }
      __syncthreads();
#pragma unroll
      for (int kk = 0; kk < 32; kk += 4) {
        const v2f a = *(const v2f*)&Hk[wave * 16 + l16][kk + khi];
        const v2f b = *(const v2f*)&Whs[l16][k0 + kk + khi];
        acc = wmma_f32_k4(a, b, acc);
      }
      __syncthreads();
    }

#pragma unroll
    for (int i = 0; i < 8; ++i) {
      const int row = wave * 16 + ((lane < 16) ? i : (8 + i));  // batch index
      const float p = pre[((size_t)row * SEQ + tt) * HID + col];
      const float h = tanhf(acc[i] + p + bias);
      Hnext[row * HID + col] = h;
      out[((size_t)row * SEQ + tt) * (2 * HID) + dir * HID + col] = h;
      if (step == SEQ - 1) hn[row * HID + col] = h;
    }

    // Device-wide barrier across the NBLK blocks of this direction.
    __threadfence();
    __syncthreads();
    if (tid == 0) {
      __hip_atomic_fetch_add(&ctr[step], 1, __ATOMIC_RELEASE,
                             __HIP_MEMORY_SCOPE_AGENT);
      while (__hip_atomic_load(&ctr[step], __ATOMIC_ACQUIRE,
                               __HIP_MEMORY_SCOPE_AGENT) < NBLK) {
        __builtin_amdgcn_s_sleep(1);
      }
    }
    __syncthreads();
  }
}

// ---------------------------------------------------------------------------
extern "C" void kernel_launch(void* const* d_in, const int* in_sizes, int n_in,
                              void* d_out, int out_size, void* d_ws,
                              size_t ws_size, hipStream_t stream) {
  const float* x     = (const float*)d_in[0];
  const float* wx_f0 = (const float*)d_in[1];
  const float* bx_f0 = (const float*)d_in[2];
  const float* wh_f0 = (const float*)d_in[3];
  const float* bh_f0 = (const float*)d_in[4];
  const float* wx_b0 = (const float*)d_in[5];
  const float* bx_b0 = (const float*)d_in[6];
  const float* wh_b0 = (const float*)d_in[7];
  const float* bh_b0 = (const float*)d_in[8];
  const float* wx_f1 = (const float*)d_in[9];
  const float* bx_f1 = (const float*)d_in[10];
  const float* wh_f1 = (const float*)d_in[11];
  const float* bh_f1 = (const float*)d_in[12];
  const float* wx_b1 = (const float*)d_in[13];
  const float* bx_b1 = (const float*)d_in[14];
  const float* wh_b1 = (const float*)d_in[15];
  const float* bh_b1 = (const float*)d_in[16];

  // Workspace layout (~256.5 MiB)
  float* ws    = (float*)d_ws;
  float* out0  = ws;                                        // [B][S][2H]
  float* preF  = out0 + (size_t)BATCH * SEQ * 2 * HID;      // [B][S][H]
  float* preB  = preF + (size_t)BATCH * SEQ * HID;          // [B][S][H]
  float* HbufF = preB + (size_t)BATCH * SEQ * HID;          // [2][B][H]
  float* HbufB = HbufF + (size_t)2 * BATCH * HID;
  int*   ctr   = (int*)(HbufB + (size_t)2 * BATCH * HID);   // [2][S]
  int*   ctrF  = ctr;
  int*   ctrB  = ctr + SEQ;

  float* out1 = (float*)d_out;                              // [B][S][2H]
  float* hn   = out1 + (size_t)BATCH * SEQ * 2 * HID;       // [4][B][H]
  float* hnF0 = hn;
  float* hnB0 = hn + (size_t)BATCH * HID;
  float* hnF1 = hn + (size_t)2 * BATCH * HID;
  float* hnB1 = hn + (size_t)3 * BATCH * HID;

  const int M = BATCH * SEQ;      // 32768
  dim3 gproj(HID / 64, M / 64);   // (8, 512)

  // ---- Layer 0: pre = x @ Wx^T + bx (K = 512) ----
  proj_gemm<<<gproj, 256, 0, stream>>>(x, wx_f0, bx_f0, preF, M, HID, HID);
  proj_gemm<<<gproj, 256, 0, stream>>>(x, wx_b0, bx_b0, preB, M, HID, HID);
  init_state<<<(2 * BATCH * HID + 255) / 256, 256, 0, stream>>>(HbufF, HbufB, ctr);
  rnn_scan<<<2 * NBLK, 128, 0, stream>>>(preF, preB, wh_f0, wh_b0, bh_f0, bh_b0,
                                         out0, HbufF, HbufB, ctrF, ctrB,
                                         hnF0, hnB0);

  // ---- Layer 1: pre = out0 @ Wx^T + bx (K = 1024) ----
  proj_gemm<<<gproj, 256, 0, stream>>>(out0, wx_f1, bx_f1, preF, M, HID, 2 * HID);
  proj_gemm<<<gproj, 256, 0, stream>>>(out0, wx_b1, bx_b1, preB, M, HID, 2 * HID);
  init_state<<<(2 * BATCH * HID + 255) / 256, 256, 0, stream>>>(HbufF, HbufB, ctr);
  rnn_scan<<<2 * NBLK, 128, 0, stream>>>(preF, preB, wh_f1, wh_b1, bh_f1, bh_b1,
                                         out1, HbufF, HbufB, ctrF, ctrB,
                                         hnF1, hnB1);
}